// NoSharingGraphConv_87428354277902
// MI455X (gfx1250) — compile-verified
//
#include <hip/hip_runtime.h>
#include <cmath>

#define B_    8
#define N_    64
#define FIN_  128
#define FOUT_ 128
#define E_    8

typedef __attribute__((ext_vector_type(2))) float v2f;
typedef __attribute__((ext_vector_type(8))) float v8f;

// ---------------------------------------------------------------------------
// Kernel A: MoE routing + conditional symmetric adjacency.
// grid (h, b), 128 threads. Writes adjb[b][h][w] (8*64*64 floats) to ws.
// ---------------------------------------------------------------------------
__global__ void cond_adj_kernel(const float* __restrict__ x,
                                const float* __restrict__ adj,
                                const float* __restrict__ experts,
                                const float* __restrict__ fc_w,
                                const float* __restrict__ fc_b,
                                float* __restrict__ adjb) {
  const int h = blockIdx.x;
  const int b = blockIdx.y;
  const int t = threadIdx.x;
  __shared__ float c_sh[FIN_];
  __shared__ float rw_sh[E_];

  // c[b,f] = mean over nodes (coalesced: 128 threads sweep rows)
  float s = 0.f;
  for (int n = 0; n < N_; ++n) s += x[(b * N_ + n) * FIN_ + t];
  c_sh[t] = s * (1.0f / N_);
  __syncthreads();

  // r_w[b,e] = sigmoid(c . fc_w[e] + fc_b[e])
  if (t < E_) {
    float d = fc_b[t];
    for (int f = 0; f < FIN_; ++f) d += c_sh[f] * fc_w[t * FIN_ + f];
    rw_sh[t] = 1.0f / (1.0f + __expf(-d));
  }
  __syncthreads();

  // adj_b[b,h,w] = 0.5*((adj[h,w]+cond_e[b,h,w]) + (adj[w,h]+cond_e[b,w,h]))
  if (t < N_) {
    const int w = t;
    float ce = 0.f, cet = 0.f;
    for (int e = 0; e < E_; ++e) {
      const float r = rw_sh[e];
      ce  += r * experts[(e * N_ + h) * N_ + w];
      cet += r * experts[(e * N_ + w) * N_ + h];
    }
    adjb[(b * N_ + h) * N_ + w] =
        0.5f * ((adj[h * N_ + w] + ce) + (adj[w * N_ + h] + cet));
  }
}

// ---------------------------------------------------------------------------
// Kernel B: the W-streaming GEMM.
//   out[b,w,m] = sum_{h,n} (adjb[b,h,w]*x[b,h,n]) * W[h,w,n,m]
// grid (w=64, hs=8 K-splits of 8 h each); 256 threads = 8 wave32.
// Wave i computes FOUT tile [16i, 16i+16) with V_WMMA_F32_16X16X4_F32,
// M = batch (8 valid of 16), K = 128*8 per block. Each W element is loaded
// by exactly one lane exactly once (single-pass 256 MB stream).
// Unroll 8 -> ~24 loads outstanding per wave for HBM saturation.
// Partials -> ws (deterministic fixed-order reduce in kernel C).
// ---------------------------------------------------------------------------
__global__ void gemm_kernel(const float* __restrict__ x,
                            const float* __restrict__ W,
                            const float* __restrict__ adjb,
                            float* __restrict__ partial) {
  const int w    = blockIdx.x;          // target node (output column block)
  const int hs   = blockIdx.y;          // K-split: h in [8*hs, 8*hs+8)
  const int wave = threadIdx.x >> 5;
  const int lane = threadIdx.x & 31;
  const int m0   = wave * 16;           // FOUT tile base
  const int row  = lane & 15;           // A: M index (batch); B: N index (m)
  const int koff = (lane >> 4) << 1;    // lanes 16-31 hold K+2, K+3
  const bool bvalid = row < B_;
  const int mcol = m0 + row;

  v8f acc = {};

  for (int hh = 0; hh < 8; ++hh) {
    const int h = hs * 8 + hh;
    // Per-lane adjacency scale folded into the A operand (0 pads rows 8-15).
    const float scale = bvalid ? adjb[(row * N_ + h) * N_ + w] : 0.0f;
    const float* xr = x + ((size_t)(bvalid ? row : 0) * N_ + h) * FIN_;
    const float* Wh = W + (size_t)h * (N_ * FIN_ * FOUT_)
                        + (size_t)w * (FIN_ * FOUT_);
#pragma unroll 8
    for (int n0 = 0; n0 < FIN_; n0 += 4) {
      const int nA = n0 + koff;
      v2f a, bm;
      a.x  = scale * xr[nA];
      a.y  = scale * xr[nA + 1];
      bm.x = Wh[(size_t)nA * FOUT_ + mcol];        // 64B contig per half-wave
      bm.y = Wh[(size_t)(nA + 1) * FOUT_ + mcol];
      acc = __builtin_amdgcn_wmma_f32_16x16x4_f32(
          /*neg_a=*/false, a, /*neg_b=*/false, bm,
          /*c_mod=*/(short)0, acc, /*reuse_a=*/false, /*reuse_b=*/false);
    }
  }

  // C/D layout: lanes 0-15, VGPR v -> (M=v, N=lane). M=b (0..7 valid).
  if (lane < 16) {
    float* p = partial + (size_t)(w * 8 + hs) * B_ * FOUT_;
#pragma unroll
    for (int v = 0; v < 8; ++v) p[v * FOUT_ + m0 + lane] = acc[v];
  }
}

// ---------------------------------------------------------------------------
// Kernel C: fixed-order split-K reduction + bias. grid (w, b), 128 threads.
// ---------------------------------------------------------------------------
__global__ void reduce_kernel(const float* __restrict__ partial,
                              const float* __restrict__ bias,
                              float* __restrict__ out) {
  const int w = blockIdx.x;
  const int b = blockIdx.y;
  const int m = threadIdx.x;
  float s = bias[m];
#pragma unroll
  for (int hs = 0; hs < 8; ++hs)
    s += partial[((size_t)(w * 8 + hs) * B_ + b) * FOUT_ + m];
  out[((size_t)b * N_ + w) * FOUT_ + m] = s;
}

// ---------------------------------------------------------------------------
extern "C" void kernel_launch(void* const* d_in, const int* in_sizes, int n_in,
                              void* d_out, int out_size, void* d_ws,
                              size_t ws_size, hipStream_t stream) {
  const float* x       = (const float*)d_in[0];
  const float* adj     = (const float*)d_in[1];
  const float* W       = (const float*)d_in[2];
  const float* bias    = (const float*)d_in[3];
  const float* experts = (const float*)d_in[4];
  const float* fc_w    = (const float*)d_in[5];
  const float* fc_b    = (const float*)d_in[6];
  float* out = (float*)d_out;

  float* adjb    = (float*)d_ws;            // 8*64*64       = 128 KB
  float* partial = adjb + B_ * N_ * N_;     // 64*8*8*128    =   2 MB

  cond_adj_kernel<<<dim3(N_, B_), FIN_, 0, stream>>>(x, adj, experts, fc_w,
                                                     fc_b, adjb);
  gemm_kernel<<<dim3(N_, 8), 256, 0, stream>>>(x, W, adjb, partial);
  reduce_kernel<<<dim3(N_, B_), FOUT_, 0, stream>>>(partial, bias, out);
}